// tf_16947940950800
// MI455X (gfx1250) — compile-verified
//
#include <hip/hip_runtime.h>

typedef float v2f __attribute__((ext_vector_type(2)));
typedef float v4f __attribute__((ext_vector_type(4)));
typedef float v8f __attribute__((ext_vector_type(8)));

// ---------------------------------------------------------------------------
// Stage 1: o[b,k] = sum_ij a[b,i] * W[k,i,j] * c[b,j] + bias[k], column 63 = 1
// GEMM view: C[512x64] = U[512x4096] x V[4096x64], U[b,ij]=a[b,i]*c[b,j],
//            V[ij,k]=W[k,ij].  One 16x16 tile per wave-set; K=4096 split over
// 8 waves (uniform 128 iterations each), LDS-reduced.
// V_WMMA_F32_16X16X4_F32 (f32 A/B, f32 acc) -> exact f32 math.
// The k=63 "ones" column needs no masking: its WMMA output column is garbage
// from W row 62 (clamped, in-bounds) and is overwritten with 1.0 at write-out.
// ---------------------------------------------------------------------------
__global__ __launch_bounds__(256) void bilinear_wmma(
    const float* __restrict__ x, const float* __restrict__ y,
    const float* __restrict__ z,
    const float* __restrict__ W1, const float* __restrict__ b1,
    const float* __restrict__ W2, const float* __restrict__ b2,
    const float* __restrict__ W3, const float* __restrict__ b3,
    float* __restrict__ o123 /* [3][512][64] */) {
  __shared__ float xs[16][64];
  __shared__ float ys[16][64];
  __shared__ float red[8][16][16];

  const int which = blockIdx.z;
  const float *A_, *C_, *W, *bias;
  if (which == 0)      { A_ = x; C_ = y; W = W1; bias = b1; }
  else if (which == 1) { A_ = x; C_ = z; W = W2; bias = b2; }
  else                 { A_ = y; C_ = z; W = W3; bias = b3; }

  const int b0 = blockIdx.x << 4;   // batch tile base
  const int n0 = blockIdx.y << 4;   // output-column tile base
  const int t  = threadIdx.x;

  // Stage the 16 a-rows and c-rows (16x64 each) into LDS.
  for (int e = t; e < 16 * 64; e += 256) {
    const int m = e >> 6, c = e & 63;
    xs[m][c] = A_[(b0 + m) * 64 + c];
    ys[m][c] = C_[(b0 + m) * 64 + c];
  }
  __syncthreads();

  const int lane = t & 31;
  const int wave = t >> 5;
  const int m    = lane & 15;        // A: M row / B: N col (per ISA layout)
  const int chi  = (lane >> 4) << 1; // K half: lanes 0-15 -> {0,1}, 16-31 -> {2,3}

  // B-matrix source: row n of W, read transposed. Clamp keeps loads in bounds;
  // the resulting column is discarded at write-out (ones column).
  const int gn   = n0 + m;
  const int wrow = (gn < 63) ? gn : 62;
  const float* __restrict__ wr = W + (size_t)wrow * 4096;

  v8f acc = {0.f, 0.f, 0.f, 0.f, 0.f, 0.f, 0.f, 0.f};

  // Wave w covers kk = w*4 + it*32; uniform 128-iteration trip count.
#pragma unroll 4
  for (int it = 0; it < 128; ++it) {
    const int kk  = (wave << 2) + (it << 5);
    const int ij0 = kk + chi;
    const int ij1 = ij0 + 1;
    v2f a, b;
    a.x = xs[m][ij0 >> 6] * ys[m][ij0 & 63];
    a.y = xs[m][ij1 >> 6] * ys[m][ij1 & 63];
    b.x = wr[ij0];
    b.y = wr[ij1];
    acc = __builtin_amdgcn_wmma_f32_16x16x4_f32(
        /*neg_a=*/false, a, /*neg_b=*/false, b,
        /*c_mod=*/(short)0, acc, /*reuse_a=*/false, /*reuse_b=*/false);
  }

  // Spill each wave's 16x16 partial C tile (ISA layout: VGPR v -> M=v or v+8).
  const int mbase = (lane >> 4) << 3;
#pragma unroll
  for (int v = 0; v < 8; ++v) red[wave][mbase + v][m] = acc[v];
  __syncthreads();

  // 256 threads: one (m,n) element each; sum 8 waves, add bias / ones column.
  const int mm = t >> 4, nn = t & 15;
  float s = 0.f;
#pragma unroll
  for (int w = 0; w < 8; ++w) s += red[w][mm][nn];
  const int col = n0 + nn;
  const float val = (col < 63) ? (s + bias[col]) : 1.0f;
  o123[((size_t)which << 15) + (size_t)(b0 + mm) * 64 + col] = val;
}

// ---------------------------------------------------------------------------
// Stage 2: out[b, ((i*64+j)*64+k)] = o1[b,i]*o2[b,j]*o3[b,k]
// One block per (b,i) slab = 4096 contiguous floats = 16 KB.
// Each thread: 4 non-temporal v4f stores; store q across the block covers a
// contiguous 4 KB run -> pure streaming writes at HBM bandwidth.
// ---------------------------------------------------------------------------
__global__ __launch_bounds__(256) void outer_kernel(
    const float* __restrict__ o1, const float* __restrict__ o2,
    const float* __restrict__ o3, float* __restrict__ out) {
  const int blk = blockIdx.x;     // 0..32767
  const int b   = blk >> 6;
  const int i   = blk & 63;
  const int t   = threadIdx.x;    // 0..255

  const float v1 = o1[b * 64 + i];
  const v4f* __restrict__ o3v = (const v4f*)(o3 + b * 64);
  const v4f c3 = o3v[t & 15];     // k = (t%16)*4 .. +3

  v4f* __restrict__ base = (v4f*)(out + ((size_t)blk << 12));

#pragma unroll
  for (int q = 0; q < 4; ++q) {
    const int j = (q << 4) + (t >> 4);
    const float s = v1 * o2[b * 64 + j];
    v4f r;
    r.x = s * c3.x; r.y = s * c3.y; r.z = s * c3.z; r.w = s * c3.w;
    // element offset q*1024 + t*4  ==  (j*64 + k) for this (j, k-quad)
    __builtin_nontemporal_store(r, base + (q << 8) + t);
  }
}

// ---------------------------------------------------------------------------
extern "C" void kernel_launch(void* const* d_in, const int* in_sizes, int n_in,
                              void* d_out, int out_size, void* d_ws,
                              size_t ws_size, hipStream_t stream) {
  const float* x  = (const float*)d_in[0];
  const float* y  = (const float*)d_in[1];
  const float* z  = (const float*)d_in[2];
  const float* W1 = (const float*)d_in[3];
  const float* b1 = (const float*)d_in[4];
  const float* W2 = (const float*)d_in[5];
  const float* b2 = (const float*)d_in[6];
  const float* W3 = (const float*)d_in[7];
  const float* b3 = (const float*)d_in[8];

  float* o = (float*)d_ws;              // [3][512][64] = 384 KB scratch
  float* out = (float*)d_out;           // [512][262144]

  dim3 g1(32, 4, 3);                    // batch tiles x n tiles x 3 bilinears
  bilinear_wmma<<<g1, 256, 0, stream>>>(x, y, z, W1, b1, W2, b2, W3, b3, o);

  const float* o1 = o;
  const float* o2 = o + 512 * 64;
  const float* o3 = o + 2 * 512 * 64;
  outer_kernel<<<32768, 256, 0, stream>>>(o1, o2, o3, out);
}